// AEVComputer_68762426409288
// MI455X (gfx1250) — compile-verified
//
#include <hip/hip_runtime.h>

typedef __attribute__((ext_vector_type(2))) float v2f;
typedef __attribute__((ext_vector_type(8))) float v8f;

#define PI_F 3.14159265358979323846f

// cos/sin of SHF_Z = pi/16 * (1,3,5,...,15)
__device__ __constant__ float COSZ[8] = {
     0.98078528f,  0.83146961f,  0.55557023f,  0.19509032f,
    -0.19509032f, -0.55557023f, -0.83146961f, -0.98078528f };
__device__ __constant__ float SINZ[8] = {
     0.19509032f,  0.55557023f,  0.83146961f,  0.98078528f,
     0.98078528f,  0.83146961f,  0.55557023f,  0.19509032f };
// TRIU pair index table for 4 species (symmetric upper-triangular numbering)
__device__ __constant__ int PAIRIDX[16] = {
    0, 1, 2, 3,
    1, 4, 5, 6,
    2, 5, 7, 8,
    3, 6, 8, 9 };

// One block per molecule (N=16 blocks, A=64 atoms, 256 threads = 8 wave32).
__global__ __launch_bounds__(256) void aev_kernel(
    const int*   __restrict__ species,
    const float* __restrict__ coords,
    float*       __restrict__ out)
{
    const int n    = blockIdx.x;
    const int t    = threadIdx.x;
    const int lane = t & 31;
    const int wave = t >> 5;

    __shared__ float sX[64][4];     // centered coords, padded K=4 (w = 0)
    __shared__ float sG[64][64];    // Gram matrix X*X^T
    __shared__ float sNorm[64];     // |x_i|^2
    __shared__ int   sSpec[64];
    __shared__ int   sNbr[64][64];  // per-atom angular neighbor list (d <= 3.5)
    __shared__ int   sCnt[64];
    __shared__ float sMean[4];

    // ---- load molecule into LDS; emit species output (value-cast to f32) ----
    if (t < 64) {
        const int gi = n * 64 + t;
        sX[t][0] = coords[gi * 3 + 0];
        sX[t][1] = coords[gi * 3 + 1];
        sX[t][2] = coords[gi * 3 + 2];
        sX[t][3] = 0.0f;
        const int sp = species[gi];
        sSpec[t] = sp;
        out[gi] = (float)sp;        // first output of the tuple
    }
    __syncthreads();

    // ---- center coordinates on the molecule mean (translation-invariant;
    //      shrinks |x|^2 ~4x -> less fp32 cancellation in d^2 = ni+nj-2Gij) ----
    if (t < 3) {
        float s = 0.0f;
        for (int j = 0; j < 64; ++j) s += sX[j][t];
        sMean[t] = s * (1.0f / 64.0f);
    }
    __syncthreads();
    if (t < 64) {
        sX[t][0] -= sMean[0];
        sX[t][1] -= sMean[1];
        sX[t][2] -= sMean[2];
    }
    __syncthreads();

    // ---- Phase 1: Gram matrix via V_WMMA_F32_16X16X4_F32 ----
    // A (16x4 f32) layout: lanes 0-15 hold K=0,1 of row M=lane; lanes 16-31 hold K=2,3.
    // B (4x16)  mirrors with N=lane&15.  D: VGPR r, lanes<16 -> M=r, lanes>=16 -> M=r+8, N=lane&15.
    {
        const int m     = lane & 15;
        const int khalf = lane >> 4;          // 0 -> K=0,1 ; 1 -> K=2,3
        #pragma unroll
        for (int it = 0; it < 2; ++it) {
            const int tile = wave * 2 + it;   // 16 tiles of the 64x64 Gram
            const int bi = tile >> 2;
            const int bj = tile & 3;
            v2f a, b;
            const int rowA = bi * 16 + m;
            const int rowB = bj * 16 + m;
            a.x = sX[rowA][khalf * 2 + 0];
            a.y = sX[rowA][khalf * 2 + 1];    // K=3 column is the zero pad
            b.x = sX[rowB][khalf * 2 + 0];
            b.y = sX[rowB][khalf * 2 + 1];
            v8f c = {};
            v8f d = __builtin_amdgcn_wmma_f32_16x16x4_f32(
                false, a, false, b, (short)0, c, false, false);
            #pragma unroll
            for (int r = 0; r < 8; ++r) {
                const int M = bi * 16 + r + khalf * 8;
                const int N = bj * 16 + m;
                sG[M][N] = d[r];
            }
        }
    }
    __syncthreads();
    if (t < 64) sNorm[t] = sG[t][t];
    __syncthreads();

    // ---- build per-atom angular neighbor lists (d^2 <= 3.5^2), one thread/atom,
    //      serial compaction -> no atomics, no divergence between atoms ----
    if (t < 64) {
        int cnt = 0;
        const float ni = sNorm[t];
        for (int j = 0; j < 64; ++j) {
            const float d2 = ni + sNorm[j] - 2.0f * sG[t][j];
            if (j != t && d2 <= 12.25f) sNbr[t][cnt++] = j;
        }
        sCnt[t] = cnt;
    }
    __syncthreads();

    float* const outAev = out + 16 * 64;      // aev output base (after species)

    // ---- Phase 2: radial AEV. thread = (atom i, shift-quarter rq) ----
    {
        const int i  = t >> 2;
        const int rq = t & 3;
        const float ni = sNorm[i];
        float acc[4][4] = {};                 // [species][shift-in-quarter]
        for (int j = 0; j < 64; ++j) {
            if (j == i) continue;
            const float d2 = ni + sNorm[j] - 2.0f * sG[i][j];
            if (d2 > 27.04f) continue;        // RCR^2 early-out, skips the sqrt too
            const float dij = sqrtf(fmaxf(d2, 0.0f));
            const float fc = 0.5f * __cosf(PI_F * dij * (1.0f / 5.2f)) + 0.5f;
            const int   sj = sSpec[j];
            #pragma unroll
            for (int mm = 0; mm < 4; ++mm) {
                const float shf  = 0.9f + 0.26875f * (float)(rq * 4 + mm);
                const float dd   = dij - shf;
                const float term = 0.25f * __expf(-16.0f * dd * dd) * fc;
                #pragma unroll
                for (int ss = 0; ss < 4; ++ss)
                    acc[ss][mm] += (sj == ss) ? term : 0.0f;   // one-hot scatter
            }
        }
        float* o = outAev + (n * 64 + i) * 384;
        #pragma unroll
        for (int ss = 0; ss < 4; ++ss)
            #pragma unroll
            for (int mm = 0; mm < 4; ++mm)
                o[ss * 16 + rq * 4 + mm] = acc[ss][mm];
    }

    // ---- Phase 3: angular AEV. wave = atom i, lane = (shf_a, shf_z) feature.
    //      Iterate only real neighbor pairs (~15-20) instead of all 2016 (j,k). ----
    {
        const int   aidx = lane >> 3;
        const int   zidx = lane & 7;
        const float shfa = 0.9f + 0.65f * (float)aidx;
        const float cz = COSZ[zidx];
        const float sz = SINZ[zidx];
        for (int i = wave; i < 64; i += 8) {
            const float ni  = sNorm[i];
            const float gii = sG[i][i];
            const int   cnt = sCnt[i];
            float acc[10] = {};
            for (int jj = 0; jj < cnt; ++jj) {
                const int   j   = sNbr[i][jj];
                const float gij = sG[i][j];
                const float dij = sqrtf(fmaxf(ni + sNorm[j] - 2.0f * gij, 0.0f));
                const float fcj = 0.5f * __cosf(PI_F * dij * (1.0f / 3.5f)) + 0.5f;
                const int   sj  = sSpec[j];
                for (int kk = jj + 1; kk < cnt; ++kk) {   // unordered pairs (x2 * 0.5)
                    const int   k   = sNbr[i][kk];
                    const float gik = sG[i][k];
                    const float dik = sqrtf(fmaxf(ni + sNorm[k] - 2.0f * gik, 0.0f));
                    const float fck = 0.5f * __cosf(PI_F * dik * (1.0f / 3.5f)) + 0.5f;
                    // dot(x_j - x_i, x_k - x_i) from the Gram matrix
                    const float dot = sG[j][k] - gij - gik + gii;
                    float c = 0.95f * dot / (dij * dik);
                    c = fminf(fmaxf(c, -0.95f), 0.95f);
                    const float s    = sqrtf(1.0f - c * c);      // sin(theta) >= 0
                    const float cosd = c * cz + s * sz;          // cos(theta - shf_z)
                    const float x    = 0.5f * (1.0f + cosd);
                    const float x2 = x * x, x4 = x2 * x2, x8 = x4 * x4, x16 = x8 * x8;
                    const float f1 = x16 * x16;                  // x^32 (ZETA)
                    const float da = 0.5f * (dij + dik) - shfa;
                    const float f2 = __expf(-8.0f * da * da);
                    const float term = 2.0f * fcj * fck * f2 * f1;
                    const int p = PAIRIDX[sj * 4 + sSpec[k]];
                    #pragma unroll
                    for (int pp = 0; pp < 10; ++pp)
                        acc[pp] += (pp == p) ? term : 0.0f;      // pair one-hot scatter
                }
            }
            float* o = outAev + (n * 64 + i) * 384 + 64;
            #pragma unroll
            for (int pp = 0; pp < 10; ++pp)
                o[pp * 32 + lane] = acc[pp];
        }
    }
}

extern "C" void kernel_launch(void* const* d_in, const int* in_sizes, int n_in,
                              void* d_out, int out_size, void* d_ws, size_t ws_size,
                              hipStream_t stream) {
    const int*   species = (const int*)d_in[0];
    const float* coords  = (const float*)d_in[1];
    float*       out     = (float*)d_out;
    (void)in_sizes; (void)n_in; (void)out_size; (void)d_ws; (void)ws_size;
    aev_kernel<<<16, 256, 0, stream>>>(species, coords, out);
}